// SAGEModel_45226005627219
// MI455X (gfx1250) — compile-verified
//
#include <hip/hip_runtime.h>
#include <hip/hip_bf16.h>

#define NN 100000
#define NE 800000
#define DF 128
#define NCLS 64
#define LN_EPS 1e-5f
#define TSUB 5                      // node sub-tiles (of 16) per block: 80 nodes, 100000/80 = 1250

typedef __attribute__((ext_vector_type(2))) float v2f;
typedef __attribute__((ext_vector_type(8))) float v8f;

// ---------------- utility kernels ----------------

__global__ void sage_zero_f32(float* __restrict__ p, long n) {
    long i = (long)blockIdx.x * blockDim.x + threadIdx.x;
    long stride = (long)gridDim.x * blockDim.x;
    for (; i < n; i += stride) p[i] = 0.0f;
}

__global__ void sage_degree(const int* __restrict__ dst, float* __restrict__ deg) {
    int e = blockIdx.x * blockDim.x + threadIdx.x;
    if (e < NE) atomicAdd(&deg[dst[e]], 1.0f);
}

__global__ void sage_invdeg(float* __restrict__ deg) {
    int i = blockIdx.x * blockDim.x + threadIdx.x;
    if (i < NN) deg[i] = 1.0f / fmaxf(deg[i], 1.0f);
}

// one wave per edge: gather 128-float row of h[src], atomically add into agg[dst]
__global__ __launch_bounds__(256) void sage_aggregate(const float* __restrict__ h,
                                                      const int* __restrict__ dst,
                                                      const int* __restrict__ src,
                                                      float* __restrict__ agg) {
    int warp = (int)(((long)blockIdx.x * blockDim.x + threadIdx.x) >> 5);
    if (warp >= NE) return;
    int lane = threadIdx.x & 31;
    int s = src[warp];
    int d = dst[warp];
    const float4 v = ((const float4*)(h + (size_t)s * DF))[lane];
    float* ap = agg + (size_t)d * DF + lane * 4;
    atomicAdd(ap + 0, v.x);
    atomicAdd(ap + 1, v.y);
    atomicAdd(ap + 2, v.z);
    atomicAdd(ap + 3, v.w);
}

// ---------------- fused layer: z = (agg*inv_deg)@Wl^T + bl + h@Wr^T ; LN ; ReLU ----------------
// block = 256 threads (8 waves). Block handles 80 nodes (5 sub-tiles of 16);
// wave w handles output features [16w, 16w+16) for ALL sub-tiles, so each B-fragment
// (Wl/Wr) load is reused by 5 WMMAs -> 12 b64 loads per K-step feed 10 WMMAs.
// WMMA f32 16x16x4: A lane layout M=lane%16, K=kk+2*(lane/16)+{0,1}; B mirrors with N=lane%16.
// D: lane holds feature j, VGPR r holds node row r + 8*(lane/16).
__global__ __launch_bounds__(256) void sage_layer(const float* __restrict__ agg,
                                                  const float* __restrict__ inv_deg,
                                                  const float* __restrict__ h,
                                                  const float* __restrict__ Wl,
                                                  const float* __restrict__ Wr,
                                                  const float* __restrict__ bl,
                                                  const float* __restrict__ gamma,
                                                  const float* __restrict__ beta,
                                                  float* __restrict__ out) {
    __shared__ float s_sum[16 * TSUB];
    __shared__ float s_sq[16 * TSUB];

    const int lane = threadIdx.x & 31;
    const int wave = threadIdx.x >> 5;   // 0..7
    const int half = lane >> 4;          // 0 or 1
    const int l16  = lane & 15;
    const int n0   = blockIdx.x * (16 * TSUB);
    const int row  = n0 + l16;           // A-matrix row (sub-tile 0) for this lane
    const int j    = wave * 16 + l16;    // output feature (B column) this lane owns

    if (threadIdx.x < 16 * TSUB) { s_sum[threadIdx.x] = 0.0f; s_sq[threadIdx.x] = 0.0f; }
    __syncthreads();

    float scale[TSUB];
#pragma unroll
    for (int t = 0; t < TSUB; ++t) scale[t] = inv_deg[row + t * 16];

    const float* aggp = agg + (size_t)row * DF;
    const float* hp   = h   + (size_t)row * DF;
    const float* wlp  = Wl  + (size_t)j * DF;
    const float* wrp  = Wr  + (size_t)j * DF;

    v8f acc[TSUB] = {};
#pragma unroll 4
    for (int kk = 0; kk < DF; kk += 4) {
        const int kq = kk + 2 * half;
        const v2f b1 = *(const v2f*)(wlp + kq);   // Wl^T fragment, shared by all sub-tiles
        const v2f b2 = *(const v2f*)(wrp + kq);   // Wr^T fragment, shared by all sub-tiles
#pragma unroll
        for (int t = 0; t < TSUB; ++t) {
            v2f a1 = *(const v2f*)(aggp + (size_t)t * 16 * DF + kq);
            a1 = a1 * scale[t];
            const v2f a2 = *(const v2f*)(hp + (size_t)t * 16 * DF + kq);
            acc[t] = __builtin_amdgcn_wmma_f32_16x16x4_f32(false, a1, false, b1, (short)0, acc[t], false, false);
            acc[t] = __builtin_amdgcn_wmma_f32_16x16x4_f32(false, a2, false, b2, (short)0, acc[t], false, false);
        }
    }

    const float bj = bl[j];
    const float gj = gamma[j];
    const float bt = beta[j];

    // per-node LayerNorm statistics: reduce over 16-lane feature groups, then across 8 waves via LDS
#pragma unroll
    for (int t = 0; t < TSUB; ++t) {
#pragma unroll
        for (int r = 0; r < 8; ++r) {
            float z = acc[t][r] + bj;
            acc[t][r] = z;
            float s = z;
            float q = z * z;
#pragma unroll
            for (int m = 8; m >= 1; m >>= 1) {
                s += __shfl_xor(s, m, 16);
                q += __shfl_xor(q, m, 16);
            }
            if (l16 == 0) {
                atomicAdd(&s_sum[t * 16 + r + 8 * half], s);
                atomicAdd(&s_sq[t * 16 + r + 8 * half], q);
            }
        }
    }
    __syncthreads();

#pragma unroll
    for (int t = 0; t < TSUB; ++t) {
#pragma unroll
        for (int r = 0; r < 8; ++r) {
            const int nloc = t * 16 + r + 8 * half;
            const float mu  = s_sum[nloc] * (1.0f / 128.0f);
            const float var = s_sq[nloc] * (1.0f / 128.0f) - mu * mu;
            const float rs  = rsqrtf(var + LN_EPS);
            float y = (acc[t][r] - mu) * rs * gj + bt;
            y = fmaxf(y, 0.0f);
            out[(size_t)(n0 + nloc) * DF + j] = y;
        }
    }
}

// ---------------- final: logits = h@Wf^T + bf ; log_softmax over 64 classes ----------------
// block = 128 threads (4 waves). Block handles 16 nodes; wave w handles classes [16w,16w+16).
__global__ __launch_bounds__(128) void sage_final(const float* __restrict__ h,
                                                  const float* __restrict__ Wf,
                                                  const float* __restrict__ bf,
                                                  float* __restrict__ out) {
    __shared__ float s_max[4][16];
    __shared__ float s_sum[4][16];

    const int lane = threadIdx.x & 31;
    const int wave = threadIdx.x >> 5;   // 0..3
    const int half = lane >> 4;
    const int l16  = lane & 15;
    const int n0   = blockIdx.x * 16;
    const int row  = n0 + l16;
    const int c    = wave * 16 + l16;    // class this lane owns

    const float* hp = h  + (size_t)row * DF;
    const float* wp = Wf + (size_t)c * DF;

    v8f acc = {};
#pragma unroll 8
    for (int kk = 0; kk < DF; kk += 4) {
        const int kq = kk + 2 * half;
        const v2f a = *(const v2f*)(hp + kq);
        const v2f b = *(const v2f*)(wp + kq);
        acc = __builtin_amdgcn_wmma_f32_16x16x4_f32(false, a, false, b, (short)0, acc, false, false);
    }

    const float bc = bf[c];
    // per-node max over this wave's 16 classes
#pragma unroll
    for (int r = 0; r < 8; ++r) {
        float z = acc[r] + bc;
        acc[r] = z;
        float m = z;
#pragma unroll
        for (int k = 8; k >= 1; k >>= 1) m = fmaxf(m, __shfl_xor(m, k, 16));
        if (l16 == 0) s_max[wave][r + 8 * half] = m;
    }
    __syncthreads();

    float gmax[8];
#pragma unroll
    for (int r = 0; r < 8; ++r) {
        const int nloc = r + 8 * half;
        const float m = fmaxf(fmaxf(s_max[0][nloc], s_max[1][nloc]),
                              fmaxf(s_max[2][nloc], s_max[3][nloc]));
        gmax[r] = m;
        float e = expf(acc[r] - m);
        float s = e;
#pragma unroll
        for (int k = 8; k >= 1; k >>= 1) s += __shfl_xor(s, k, 16);
        if (l16 == 0) s_sum[wave][nloc] = s;
    }
    __syncthreads();

#pragma unroll
    for (int r = 0; r < 8; ++r) {
        const int nloc = r + 8 * half;
        const float tot = s_sum[0][nloc] + s_sum[1][nloc] + s_sum[2][nloc] + s_sum[3][nloc];
        const float ls = logf(tot);
        out[(size_t)(n0 + nloc) * NCLS + c] = acc[r] - gmax[r] - ls;
    }
}

// ---------------- launcher ----------------

extern "C" void kernel_launch(void* const* d_in, const int* in_sizes, int n_in,
                              void* d_out, int out_size, void* d_ws, size_t ws_size,
                              hipStream_t stream) {
    const float* x     = (const float*)d_in[0];
    const int*   ei    = (const int*)d_in[1];
    const float* Wl    = (const float*)d_in[2];
    const float* bl    = (const float*)d_in[3];
    const float* Wr    = (const float*)d_in[4];
    const float* gamma = (const float*)d_in[5];
    const float* beta  = (const float*)d_in[6];
    const float* Wf    = (const float*)d_in[7];
    const float* bf    = (const float*)d_in[8];
    float* out = (float*)d_out;

    const int* dst = ei;             // edge_index[0]
    const int* src = ei + NE;        // edge_index[1]

    float* buf0 = (float*)d_ws;                       // 100000*128 f32
    float* buf1 = buf0 + (size_t)NN * DF;             // 100000*128 f32
    float* deg  = buf1 + (size_t)NN * DF;             // 100000 f32

    const long feat_elems = (long)NN * DF;

    // degree -> inv_deg (recomputed every call: deterministic, stateless)
    sage_zero_f32<<<1024, 256, 0, stream>>>(deg, (long)NN);
    sage_degree<<<(NE + 255) / 256, 256, 0, stream>>>(dst, deg);
    sage_invdeg<<<(NN + 255) / 256, 256, 0, stream>>>(deg);

    const float* hcur = x;
    float* bufs[2] = { buf0, buf1 };
    for (int l = 0; l < 3; ++l) {
        float* aggbuf = bufs[l & 1];  // L0:buf0, L1:buf1, L2:buf0 (h1 dead by then)
        sage_zero_f32<<<4096, 256, 0, stream>>>(aggbuf, feat_elems);
        sage_aggregate<<<(NE * 32 + 255) / 256, 256, 0, stream>>>(hcur, dst, src, aggbuf);
        sage_layer<<<NN / (16 * TSUB), 256, 0, stream>>>(aggbuf, deg, hcur,
                                                         Wl + (size_t)l * DF * DF,
                                                         Wr + (size_t)l * DF * DF,
                                                         bl + (size_t)l * DF,
                                                         gamma + (size_t)l * DF,
                                                         beta + (size_t)l * DF,
                                                         aggbuf /* in-place: block-local rows only */);
        hcur = aggbuf;
    }

    sage_final<<<NN / 16, 128, 0, stream>>>(hcur, Wf, bf, out);
}